// omm_37048387895735
// MI455X (gfx1250) — compile-verified
//
#include <hip/hip_runtime.h>
#include <math.h>

// Problem constants (from reference): L=50, M=50, N=1900, ITER=76, STEP=38, B=16
#define LL    50
#define BB    16
#define ITERR 76
#define KDIM  100     // non-zero rows of first FFT (2*L)
#define CDIM  3800    // non-zero cols (ITER*L)
#define CPAD  3808    // padded to multiple of 16 (238 tiles)
#define JPAD  64      // 50 -> 4 tiles of 16
#define SROW  7600    // full_sig row stride
#define SROW0 50      // first non-zero row of A in full_sig
#define COL0  1900    // first non-zero col of A in full_sig

typedef float v2f __attribute__((ext_vector_type(2)));
typedef float v8f __attribute__((ext_vector_type(8)));

// D = A(16x4 f32) * B(4x16 f32) + C(16x16 f32), wave32
#define WMMA_F32X4(a, b, c) \
  __builtin_amdgcn_wmma_f32_16x16x4_f32(false, (a), false, (b), (short)0, (c), false, false)

// ---------------------------------------------------------------------------
// K0: per-batch max(input[b]) into smax[0..15]; max|weight| into smax[16]
// ---------------------------------------------------------------------------
__global__ void k_maxes(const float* __restrict__ inp, const float* __restrict__ wgt,
                        float* __restrict__ smax) {
  __shared__ float red[256];
  int b = blockIdx.x;
  float m = -3.4e38f;
  if (b < BB) {
    const float* p = inp + b * 2500;
    for (int i = threadIdx.x; i < 2500; i += 256) m = fmaxf(m, p[i]);
  } else {
    for (int i = threadIdx.x; i < 2500; i += 256) m = fmaxf(m, fabsf(wgt[i]));
  }
  red[threadIdx.x] = m;
  __syncthreads();
  for (int s = 128; s > 0; s >>= 1) {
    if (threadIdx.x < s) red[threadIdx.x] = fmaxf(red[threadIdx.x], red[threadIdx.x + s]);
    __syncthreads();
  }
  if (threadIdx.x == 0) smax[b] = red[0];
}

// ---------------------------------------------------------------------------
// K1: build transposed 2nd-stage twiddles Qt[i][c] = exp(-i*pi*((i+1)(1900+c) mod 200)/100)
//     rows i >= 50 zeroed (tile padding)
// ---------------------------------------------------------------------------
__global__ void k_build_q(float* __restrict__ qre, float* __restrict__ qim) {
  int idx = blockIdx.x * 256 + threadIdx.x;
  if (idx >= JPAD * CPAD) return;
  int i = idx / CPAD, c = idx % CPAD;
  float re = 0.f, im = 0.f;
  if (i < LL) {
    int t = ((i + 1) * (COL0 + c)) % 200;   // exact integer phase reduction
    float s, cz;
    sincospif((float)t * 0.01f, &s, &cz);
    re = cz; im = -s;
  }
  qre[idx] = re; qim[idx] = im;
}

// ---------------------------------------------------------------------------
// K2: GEMM-1 with fused A' construction + |W| epilogue.
//   G[b,j,c] = |W[j, c/76]| * sum_r P[j,r] * S[50+r, 1900+c] * inp[b, r>>1, c/76]
//   Tiles: M = j (64 padded), N = c (3808 padded), K = r (100).
//   Grid: b(16) x jtile(4) x cgroup(60); 4 waves/block, wave w -> ctile cg*4+w.
// ---------------------------------------------------------------------------
__global__ void __launch_bounds__(128) k_mm1(
    const float* __restrict__ inp, const float* __restrict__ wgt,
    const float* __restrict__ sre, const float* __restrict__ sim,
    float* __restrict__ gre, float* __restrict__ gim) {
  __shared__ float Pre[16 * KDIM];
  __shared__ float Pim[16 * KDIM];
  int bid = blockIdx.x;
  int cg = bid % 60;
  int jt = (bid / 60) % 4;
  int b  = bid / 240;

  // Fill P twiddles for this j-tile (rows j>=50 zeroed -> G pad rows are 0)
  for (int e = threadIdx.x; e < 16 * KDIM; e += 128) {
    int jl = e / KDIM, r = e % KDIM;
    int j = jt * 16 + jl;
    float re = 0.f, im = 0.f;
    if (j < LL) {
      int t = ((150 + 2 * j) * (50 + r)) % 200;  // k1u = (150+2j) mod 200, exact
      float s, cz;
      sincospif((float)t * 0.01f, &s, &cz);
      re = cz; im = -s;
    }
    Pre[e] = re; Pim[e] = im;
  }
  __syncthreads();

  int wave = threadIdx.x >> 5;
  int lane = threadIdx.x & 31;
  int ctile = cg * 4 + wave;
  if (ctile >= CPAD / 16) return;   // wave-uniform early out (after barrier)

  int h  = lane >> 4;      // half-wave selector (K split)
  int ln = lane & 15;      // row (A) / column (B,C) within tile
  int c  = ctile * 16 + ln;
  int x2 = COL0 + c;
  int q  = (c < CDIM) ? (c / ITERR) : 0;
  const float* inp_b = inp + b * 2500 + q;

  v8f cre = {}; v8f cim = {};
  for (int t = 0; t < KDIM / 4; ++t) {
    int kk = 4 * t + 2 * h;
    // A fragment (twiddles) from LDS: row = ln, K = kk, kk+1
    v2f are = *(const v2f*)&Pre[ln * KDIM + kk];
    v2f aim = *(const v2f*)&Pim[ln * KDIM + kk];
    v2f aimn = -aim;
    // B fragment: A'[kk..kk+1][c] = S[50+kk..][1900+c] * inp[b, kk>>1, q]
    // (kk and kk+1 share the same input row since kk is even)
    float inpv = inp_b[(2 * t + h) * LL];
    int so = (SROW0 + kk) * SROW + x2;
    v2f bre; bre.x = sre[so] * inpv; bre.y = sre[so + SROW] * inpv;
    v2f bim; bim.x = sim[so] * inpv; bim.y = sim[so + SROW] * inpv;
    // complex MAC: 4 fp32 WMMAs
    cre = WMMA_F32X4(are,  bre, cre);
    cre = WMMA_F32X4(aimn, bim, cre);
    cim = WMMA_F32X4(are,  bim, cim);
    cim = WMMA_F32X4(aim,  bre, cim);
  }

  // Epilogue: multiply by |weight[j][q]|; pad rows forced to 0
  size_t gbase = (size_t)b * JPAD * CPAD;
#pragma unroll
  for (int r = 0; r < 8; ++r) {
    int j = jt * 16 + r + 8 * h;   // C layout: VGPR r -> M=r (lanes 0-15), M=r+8 (16-31)
    float w = (j < LL) ? fabsf(wgt[j * LL + q]) : 0.f;
    gre[gbase + (size_t)j * CPAD + c] = cre[r] * w;
    gim[gbase + (size_t)j * CPAD + c] = cim[r] * w;
  }
}

// ---------------------------------------------------------------------------
// K3: GEMM-2.  F[b,j,i] = sum_c G[b,j,c] * Q[c,i]  (complex, K = 3800)
//   magbuf[b][j][i] = |F| / amplitude[j,i]   (0 in pad region)
//   Grid: b(16) x jtile(4); 4 waves/block, wave w -> itile w.
// ---------------------------------------------------------------------------
__global__ void __launch_bounds__(128) k_mm2(
    const float* __restrict__ gre, const float* __restrict__ gim,
    const float* __restrict__ qre, const float* __restrict__ qim,
    const float* __restrict__ amp, float* __restrict__ magbuf) {
  int jt = blockIdx.x % 4;
  int b  = blockIdx.x / 4;
  int wave = threadIdx.x >> 5;   // itile
  int lane = threadIdx.x & 31;
  int h = lane >> 4, ln = lane & 15;
  int i = wave * 16 + ln;

  const float* gre_l = gre + (size_t)b * JPAD * CPAD + (size_t)(jt * 16 + ln) * CPAD;
  const float* gim_l = gim + (size_t)b * JPAD * CPAD + (size_t)(jt * 16 + ln) * CPAD;
  const float* qre_l = qre + (size_t)i * CPAD;   // Qt stored [i][c] -> contiguous K loads
  const float* qim_l = qim + (size_t)i * CPAD;

  v8f cre = {}; v8f cim = {};
  for (int t = 0; t < CDIM / 4; ++t) {
    int k = 4 * t + 2 * h;
    v2f are = *(const v2f*)&gre_l[k];
    v2f aim = *(const v2f*)&gim_l[k];
    v2f bre = *(const v2f*)&qre_l[k];
    v2f bim = *(const v2f*)&qim_l[k];
    v2f aimn = -aim;
    cre = WMMA_F32X4(are,  bre, cre);
    cre = WMMA_F32X4(aimn, bim, cre);
    cim = WMMA_F32X4(are,  bim, cim);
    cim = WMMA_F32X4(aim,  bre, cim);
  }

#pragma unroll
  for (int r = 0; r < 8; ++r) {
    int j = jt * 16 + r + 8 * h;
    float mag = sqrtf(cre[r] * cre[r] + cim[r] * cim[r]);
    float v = 0.f;
    if (j < LL && i < LL) v = mag / amp[j * LL + i];
    magbuf[(size_t)b * JPAD * JPAD + j * JPAD + i] = v;
  }
}

// ---------------------------------------------------------------------------
// K4: per-batch max of magbuf, then out[b,i,j] = magbuf[b,j,i]/max * (inp_max*w_max) + bias[j]
// ---------------------------------------------------------------------------
__global__ void k_final(const float* __restrict__ magbuf, const float* __restrict__ smax,
                        const float* __restrict__ bias, float* __restrict__ out) {
  __shared__ float red[256];
  int b = blockIdx.x;
  const float* mb = magbuf + (size_t)b * JPAD * JPAD;
  float m = 0.f;
  for (int idx = threadIdx.x; idx < JPAD * JPAD; idx += 256) m = fmaxf(m, mb[idx]);
  red[threadIdx.x] = m;
  __syncthreads();
  for (int s = 128; s > 0; s >>= 1) {
    if (threadIdx.x < s) red[threadIdx.x] = fmaxf(red[threadIdx.x], red[threadIdx.x + s]);
    __syncthreads();
  }
  float scale = (smax[b] * smax[16]) / red[0];
  for (int idx = threadIdx.x; idx < LL * LL; idx += 256) {
    int i = idx / LL, j = idx % LL;
    out[b * 2500 + idx] = mb[j * JPAD + i] * scale + bias[j];
  }
}

// ---------------------------------------------------------------------------
extern "C" void kernel_launch(void* const* d_in, const int* in_sizes, int n_in,
                              void* d_out, int out_size, void* d_ws, size_t ws_size,
                              hipStream_t stream) {
  const float* inp  = (const float*)d_in[0];  // (16,50,50)
  const float* wgt  = (const float*)d_in[1];  // (50,50)
  const float* bias = (const float*)d_in[2];  // (1,50)
  const float* amp  = (const float*)d_in[3];  // (1,50,50)
  const float* sre  = (const float*)d_in[4];  // (200,7600)
  const float* sim  = (const float*)d_in[5];  // (200,7600)

  float* ws   = (float*)d_ws;
  float* smax = ws;                            // 32 floats
  float* qre  = ws + 32;                       // 64*3808
  float* qim  = qre + (size_t)JPAD * CPAD;
  float* gre  = qim + (size_t)JPAD * CPAD;     // 16*64*3808
  float* gim  = gre + (size_t)BB * JPAD * CPAD;
  float* magb = gim + (size_t)BB * JPAD * CPAD; // 16*64*64

  k_maxes  <<<17, 256, 0, stream>>>(inp, wgt, smax);
  k_build_q<<<(JPAD * CPAD + 255) / 256, 256, 0, stream>>>(qre, qim);
  k_mm1    <<<BB * 4 * 60, 128, 0, stream>>>(inp, wgt, sre, sim, gre, gim);
  k_mm2    <<<BB * 4, 128, 0, stream>>>(gre, gim, qre, qim, amp, magb);
  k_final  <<<BB, 256, 0, stream>>>(magb, smax, bias, (float*)d_out);
}